// HierarchicalPolarization_8392366096431
// MI455X (gfx1250) — compile-verified
//
#include <hip/hip_runtime.h>

typedef __attribute__((ext_vector_type(2))) float v2f;
typedef __attribute__((ext_vector_type(8))) float v8f;

__device__ __forceinline__ float gelu_tanh(float x) {
    const float k0 = 0.7978845608028654f; // sqrt(2/pi)
    const float k1 = 0.044715f;
    float u = k0 * (x + k1 * x * x * x);
    return 0.5f * x * (1.0f + tanhf(u));
}

// ---------------- elementwise kernels (float4 granularity, rows are 64 float4) --------------

// A0[r,:] = 0.5*(Z[2r,:] + Z[2r+1,:])
__global__ void pair_mean_kernel(const float4* __restrict__ Z, float4* __restrict__ A0, int n4) {
    int i = blockIdx.x * blockDim.x + threadIdx.x;
    if (i >= n4) return;
    int j = ((i >> 6) << 7) + (i & 63);      // child row 2r, same column
    float4 a = Z[j], b = Z[j + 64];
    float4 o;
    o.x = 0.5f*(a.x+b.x); o.y = 0.5f*(a.y+b.y); o.z = 0.5f*(a.z+b.z); o.w = 0.5f*(a.w+b.w);
    A0[i] = o;
}

// Anext[r,:] = 0.5*((A[2r]+At[2r]) + (A[2r+1]+At[2r+1]))
__global__ void next_mean_kernel(const float4* __restrict__ A, const float4* __restrict__ At,
                                 float4* __restrict__ An, int n4) {
    int i = blockIdx.x * blockDim.x + threadIdx.x;
    if (i >= n4) return;
    int j = ((i >> 6) << 7) + (i & 63);
    float4 a0 = A[j], t0 = At[j], a1 = A[j + 64], t1 = At[j + 64];
    float4 o;
    o.x = 0.5f*(a0.x+t0.x+a1.x+t1.x);
    o.y = 0.5f*(a0.y+t0.y+a1.y+t1.y);
    o.z = 0.5f*(a0.z+t0.z+a1.z+t1.z);
    o.w = 0.5f*(a0.w+t0.w+a1.w+t1.w);
    An[i] = o;
}

// AtL[r,:] += P[r>>1,:]   (downward accumulation of corrections)
__global__ void accum_kernel(float4* __restrict__ AtL, const float4* __restrict__ P, int n4) {
    int i = blockIdx.x * blockDim.x + threadIdx.x;
    if (i >= n4) return;
    int j = ((i >> 7) << 6) + (i & 63);      // parent row r>>1, same column
    float4 a = AtL[i], p = P[j];
    a.x += p.x; a.y += p.y; a.z += p.z; a.w += p.w;
    AtL[i] = a;
}

// out[s,:] = Z[s,:] + At0[s>>1,:]
__global__ void final_add_kernel(const float4* __restrict__ Z, const float4* __restrict__ At0,
                                 float4* __restrict__ out, int n4) {
    int i = blockIdx.x * blockDim.x + threadIdx.x;
    if (i >= n4) return;
    int j = ((i >> 7) << 6) + (i & 63);
    float4 z = Z[i], p = At0[j];
    z.x += p.x; z.y += p.y; z.z += p.z; z.w += p.w;
    out[i] = z;
}

// ---------------- WMMA MLP kernel: At = (GELU(A @ W1 + b1)) @ W2 + b2 --------------------
// One wave32 per 16-row tile. f32 WMMA 16x16x4:
//   A frag (16x4):  lane&15 = row M, lane>>4 selects K pair {0,1}/{2,3}, v2f = 2 consecutive K
//   B frag (4x16):  lane&15 = col N, lane>>4 selects K pair,             v2f = 2 consecutive K
//   C/D (16x16):    vgpr i + 8*(lane>>4) = row M, lane&15 = col N
__global__ __launch_bounds__(128) void mlp_wmma_kernel(
    const float* __restrict__ A,   // (R,256)
    const float* __restrict__ W1,  // (256,32)
    const float* __restrict__ b1,  // (32)
    const float* __restrict__ W2,  // (32,256)
    const float* __restrict__ b2,  // (256)
    float* __restrict__ At,        // (R,256)
    int R)
{
    __shared__ float hbuf[4][16 * 32];     // per-wave staging of h
    const int lane = threadIdx.x & 31;
    const int wv   = threadIdx.x >> 5;
    const int tile = blockIdx.x * 4 + wv;
    const int nTiles = (R + 15) >> 4;
    if (tile >= nTiles) return;            // wave-uniform: EXEC stays all-ones

    const int col  = lane & 15;            // row for A-frags, col for B/D frags
    const int half = lane >> 4;
    const int kb   = half * 2;             // K offset within a 4-step
    const int rbase = tile * 16;
    const int arow  = min(rbase + col, R - 1);   // clamped load row (no divergence)
    const float* Arow = A + (size_t)arow * 256;

    // ---- GEMM1: h(16x32) = A(16x256) @ W1(256x32), both N-tiles in one K loop ----
    v8f acc0 = {}; v8f acc1 = {};
    for (int k0 = 0; k0 < 256; k0 += 4) {
        v2f a;  a.x  = Arow[k0 + kb];                a.y  = Arow[k0 + kb + 1];
        v2f bA; bA.x = W1[(k0 + kb) * 32 + col];     bA.y = W1[(k0 + kb + 1) * 32 + col];
        v2f bB; bB.x = W1[(k0 + kb) * 32 + 16 + col];bB.y = W1[(k0 + kb + 1) * 32 + 16 + col];
        acc0 = __builtin_amdgcn_wmma_f32_16x16x4_f32(false, a, false, bA, (short)0, acc0, false, false);
        acc1 = __builtin_amdgcn_wmma_f32_16x16x4_f32(false, a, false, bB, (short)0, acc1, false, false);
    }

    // ---- bias + GELU, stage h to LDS (LDS ops are in-order within a wave) ----
    const float bias0 = b1[col];
    const float bias1 = b1[16 + col];
    float* hs = hbuf[wv];
    #pragma unroll
    for (int i = 0; i < 8; ++i) {
        int m = i + half * 8;
        hs[m * 32 + col]      = gelu_tanh(acc0[i] + bias0);
        hs[m * 32 + 16 + col] = gelu_tanh(acc1[i] + bias1);
    }

    // ---- load h A-frags for all 8 K-steps into registers ----
    v2f af[8];
    #pragma unroll
    for (int k = 0; k < 8; ++k) {
        af[k].x = hs[col * 32 + k * 4 + kb];
        af[k].y = hs[col * 32 + k * 4 + kb + 1];
    }

    // ---- GEMM2: At(16x256) = h(16x32) @ W2(32x256) + b2, one N-tile at a time ----
    for (int nt = 0; nt < 16; ++nt) {
        const int n0 = nt * 16;
        v8f acc = {};
        #pragma unroll
        for (int k = 0; k < 8; ++k) {
            v2f b;
            b.x = W2[(k * 4 + kb) * 256 + n0 + col];
            b.y = W2[(k * 4 + kb + 1) * 256 + n0 + col];
            acc = __builtin_amdgcn_wmma_f32_16x16x4_f32(false, af[k], false, b, (short)0, acc, false, false);
        }
        const float bb = b2[n0 + col];
        #pragma unroll
        for (int i = 0; i < 8; ++i) {
            int grow = rbase + i + half * 8;
            if (grow < R)                                 // lane-predicated store only
                At[(size_t)grow * 256 + n0 + col] = acc[i] + bb;
        }
    }
}

// ------------------------------------ host launch -----------------------------------------
extern "C" void kernel_launch(void* const* d_in, const int* in_sizes, int n_in,
                              void* d_out, int out_size, void* d_ws, size_t ws_size,
                              hipStream_t stream) {
    const float* Z  = (const float*)d_in[0];   // (8,16384,256)
    const float* W1 = (const float*)d_in[1];   // (10,256,32)
    const float* b1 = (const float*)d_in[2];   // (10,32)
    const float* W2 = (const float*)d_in[3];   // (10,32,256)
    const float* b2 = (const float*)d_in[4];   // (10,256)
    float* out = (float*)d_out;

    const int ZROWS = 8 * 16384;               // 131072 flattened (b,s) rows
    const int NLEV  = 14;                      // log2(16384)

    // workspace layout (floats): A ping-pong buffers + At pyramid
    float* Abuf0  = (float*)d_ws;                          // 65536*256
    float* Abuf1  = Abuf0 + (size_t)65536 * 256;           // 32768*256
    float* Atbase = Abuf1 + (size_t)32768 * 256;
    size_t atOff[NLEV];
    {
        size_t acc = 0;
        for (int L = 0; L < NLEV; ++L) { atOff[L] = acc; acc += (size_t)(ZROWS >> (L + 1)) * 256; }
    }

    // level-0 pairwise mean of Z
    {
        int n4 = (ZROWS >> 1) * 64;            // 65536 rows * 64 float4
        pair_mean_kernel<<<(n4 + 255) / 256, 256, 0, stream>>>((const float4*)Z, (float4*)Abuf0, n4);
    }

    // upward pass: MLP per level + next-level means
    float* Acur = Abuf0;
    for (int L = 0; L < NLEV; ++L) {
        int R = ZROWS >> (L + 1);
        int t = (L < 9) ? L : 9;               // min(level, N_TRANSFORMS-1)
        float* AtL = Atbase + atOff[L];
        int tiles = (R + 15) / 16;
        mlp_wmma_kernel<<<(tiles + 3) / 4, 128, 0, stream>>>(
            Acur,
            W1 + (size_t)t * 256 * 32, b1 + (size_t)t * 32,
            W2 + (size_t)t * 32 * 256, b2 + (size_t)t * 256,
            AtL, R);
        if (L < NLEV - 1) {
            float* Anext = (Acur == Abuf0) ? Abuf1 : Abuf0;
            int n4 = (R >> 1) * 64;
            next_mean_kernel<<<(n4 + 255) / 256, 256, 0, stream>>>(
                (const float4*)Acur, (const float4*)AtL, (float4*)Anext, n4);
            Acur = Anext;
        }
    }

    // downward pass: accumulate corrections in-place (At_L += At_{L+1}[parent])
    for (int L = NLEV - 2; L >= 0; --L) {
        int R = ZROWS >> (L + 1);
        int n4 = R * 64;
        accum_kernel<<<(n4 + 255) / 256, 256, 0, stream>>>(
            (float4*)(Atbase + atOff[L]), (const float4*)(Atbase + atOff[L + 1]), n4);
    }

    // final: out = Z + broadcast(At_0)
    {
        int n4 = ZROWS * 64;
        final_add_kernel<<<(n4 + 255) / 256, 256, 0, stream>>>(
            (const float4*)Z, (const float4*)(Atbase + atOff[0]), (float4*)out, n4);
    }
}